// CBMSAGE_26087631356377
// MI455X (gfx1250) — compile-verified
//
#include <hip/hip_runtime.h>

typedef __attribute__((ext_vector_type(2))) float v2f;
typedef __attribute__((ext_vector_type(8))) float v8f;

#define D            512
#define ROWS         32      // node rows per block in fused kernel
#define LDS_STRIDE   516     // 512 + 4 pad -> conflict-free b64 LDS reads

// ---------------- CSR construction ----------------

__global__ void zero_u32_kernel(unsigned* __restrict__ p, int n) {
  int i = blockIdx.x * blockDim.x + threadIdx.x;
  if (i < n) p[i] = 0u;
}

__global__ void count_deg_kernel(const long long* __restrict__ ei,
                                 unsigned* __restrict__ deg, int E) {
  int e = blockIdx.x * blockDim.x + threadIdx.x;
  if (e < E) {
    int dst = (int)ei[(long long)E + e];
    atomicAdd(&deg[dst], 1u);
  }
}

// single-block exclusive scan (Hillis-Steele in LDS, looped)
__global__ void scan_deg_kernel(const unsigned* __restrict__ deg,
                                unsigned* __restrict__ rowptr,
                                unsigned* __restrict__ cursor, int N) {
  __shared__ unsigned sm[1024];
  __shared__ unsigned carry;
  int tid = threadIdx.x;
  if (tid == 0) carry = 0u;
  __syncthreads();
  for (int base = 0; base < N; base += 1024) {
    int i = base + tid;
    unsigned v = (i < N) ? deg[i] : 0u;
    sm[tid] = v;
    __syncthreads();
    for (int off = 1; off < 1024; off <<= 1) {
      unsigned t = (tid >= off) ? sm[tid - off] : 0u;
      __syncthreads();
      sm[tid] += t;
      __syncthreads();
    }
    unsigned excl = sm[tid] - v + carry;
    if (i < N) { rowptr[i] = excl; cursor[i] = excl; }
    __syncthreads();
    if (tid == 0) carry += sm[1023];
    __syncthreads();
  }
  if (tid == 0) rowptr[N] = carry;
}

__global__ void fill_csr_kernel(const long long* __restrict__ ei,
                                unsigned* __restrict__ cursor,
                                unsigned* __restrict__ csr, int E) {
  int e = blockIdx.x * blockDim.x + threadIdx.x;
  if (e < E) {
    int src = (int)ei[e];
    int dst = (int)ei[(long long)E + e];
    unsigned pos = atomicAdd(&cursor[dst], 1u);
    csr[pos] = (unsigned)src;
  }
}

// ---------------- fused gather + dual WMMA GEMM ----------------
// out[i,:] = mean_x[i,:] @ W_l^T + x[i,:] @ W_r^T + b_l * [deg>0]

__global__ __launch_bounds__(256)
void sage_fused_kernel(const float* __restrict__ x,
                       const float* __restrict__ Wl,
                       const float* __restrict__ bl,
                       const float* __restrict__ Wr,
                       const unsigned* __restrict__ deg,
                       const unsigned* __restrict__ rowptr,
                       const unsigned* __restrict__ csr,
                       float* __restrict__ out, int N) {
  __shared__ float sMean[ROWS][LDS_STRIDE];  // 66 KB
  __shared__ float sX[ROWS][LDS_STRIDE];     // 66 KB
  __shared__ float sFlag[ROWS];

  const int lane = threadIdx.x & 31;
  const int wave = threadIdx.x >> 5;         // 8 waves (wave32)
  const int rowBase = blockIdx.x * ROWS;

  // ---- Phase A: stage x rows + gather neighbor mean into LDS ----
  for (int r = wave; r < ROWS; r += 8) {
    int gr = rowBase + r;
    unsigned dv = 0u, begin = 0u, end = 0u;
    if (gr < N) { dv = deg[gr]; begin = rowptr[gr]; end = rowptr[gr + 1]; }
    float inv = 1.0f / (float)((dv > 0u) ? dv : 1u);
    if (lane == 0) sFlag[r] = (dv > 0u) ? 1.0f : 0.0f;

    // stage this node's own x row (b128 coalesced)
    float4* sxr = (float4*)&sX[r][0] + lane;       // row base is 16B aligned (516*4)
    if (gr < N) {
      const float4* xr = (const float4*)x + (size_t)gr * (D / 4) + lane;
      #pragma unroll
      for (int j = 0; j < 4; j++) sxr[j * 32] = xr[j * 32];
    } else {
      float4 z = make_float4(0.f, 0.f, 0.f, 0.f);
      #pragma unroll
      for (int j = 0; j < 4; j++) sxr[j * 32] = z;
    }

    // gather neighbor rows (coalesced b128 per lane, no atomics)
    float4 acc[4];
    #pragma unroll
    for (int j = 0; j < 4; j++) acc[j] = make_float4(0.f, 0.f, 0.f, 0.f);
    for (unsigned e = begin; e < end; ++e) {
      unsigned s = csr[e];
      const float4* nr = (const float4*)x + (size_t)s * (D / 4) + lane;
      if (e + 1 < end)   // prefetch next neighbor row (random-access, L2-resident x)
        __builtin_prefetch((const float4*)x + (size_t)csr[e + 1] * (D / 4) + lane, 0, 0);
      #pragma unroll
      for (int j = 0; j < 4; j++) {
        float4 t = nr[j * 32];
        acc[j].x += t.x; acc[j].y += t.y; acc[j].z += t.z; acc[j].w += t.w;
      }
    }
    float4* smr = (float4*)&sMean[r][0] + lane;
    #pragma unroll
    for (int j = 0; j < 4; j++) {
      float4 m;
      m.x = acc[j].x * inv; m.y = acc[j].y * inv;
      m.z = acc[j].z * inv; m.w = acc[j].w * inv;
      smr[j * 32] = m;
    }
  }
  __syncthreads();

  // ---- Phase B: dual GEMM fused into one accumulator via f32 WMMA ----
  const int lane15 = lane & 15;
  const int khalf  = lane >> 4;                 // K-halves {0,1}/{2,3} split
  for (int ct = wave; ct < D / 16; ct += 8) {   // 32 col tiles, 8 waves
    const int colBase = ct * 16;
    v8f c0 = {};  // rows 0..15 of this block
    v8f c1 = {};  // rows 16..31
    const float* wlp = Wl + (size_t)(colBase + lane15) * D;  // B[k][o] = W[o][k]
    const float* wrp = Wr + (size_t)(colBase + lane15) * D;

    for (int k = 0; k < D; k += 4) {            // mean @ W_l^T
      int kk = k + 2 * khalf;
      v2f a0 = *(const v2f*)&sMean[lane15][kk];
      v2f a1 = *(const v2f*)&sMean[16 + lane15][kk];
      v2f b  = *(const v2f*)&wlp[kk];
      c0 = __builtin_amdgcn_wmma_f32_16x16x4_f32(false, a0, false, b, (short)0, c0, false, false);
      c1 = __builtin_amdgcn_wmma_f32_16x16x4_f32(false, a1, false, b, (short)0, c1, false, false);
    }
    for (int k = 0; k < D; k += 4) {            // x @ W_r^T
      int kk = k + 2 * khalf;
      v2f a0 = *(const v2f*)&sX[lane15][kk];
      v2f a1 = *(const v2f*)&sX[16 + lane15][kk];
      v2f b  = *(const v2f*)&wrp[kk];
      c0 = __builtin_amdgcn_wmma_f32_16x16x4_f32(false, a0, false, b, (short)0, c0, false, false);
      c1 = __builtin_amdgcn_wmma_f32_16x16x4_f32(false, a1, false, b, (short)0, c1, false, false);
    }

    // epilogue: +bias gated by deg>0; C layout: lane=N, VGPR v -> M=v (+8 for hi lanes)
    int col = colBase + lane15;
    float bias = bl[col];
    #pragma unroll
    for (int v = 0; v < 8; v++) {
      int r0 = v + 8 * khalf;
      int r1 = 16 + r0;
      int g0 = rowBase + r0, g1 = rowBase + r1;
      if (g0 < N) out[(size_t)g0 * D + col] = c0[v] + bias * sFlag[r0];
      if (g1 < N) out[(size_t)g1 * D + col] = c1[v] + bias * sFlag[r1];
    }
  }
}

// ---------------- launcher ----------------

extern "C" void kernel_launch(void* const* d_in, const int* in_sizes, int n_in,
                              void* d_out, int out_size, void* d_ws, size_t ws_size,
                              hipStream_t stream) {
  const float*     x  = (const float*)d_in[0];
  const long long* ei = (const long long*)d_in[1];   // int64 edge_index [2,E]
  const float*     Wl = (const float*)d_in[2];
  const float*     bl = (const float*)d_in[3];
  const float*     Wr = (const float*)d_in[4];
  float* out = (float*)d_out;

  const int N = in_sizes[0] / D;
  const int E = in_sizes[1] / 2;

  // workspace layout (u32): deg[N] | rowptr[N+1] | cursor[N+1] | csr[E]  (~3.8 MB)
  unsigned* deg    = (unsigned*)d_ws;
  unsigned* rowptr = deg + N;
  unsigned* cursor = rowptr + (N + 1);
  unsigned* csr    = cursor + (N + 1);

  zero_u32_kernel <<<(N + 255) / 256, 256, 0, stream>>>(deg, N);
  count_deg_kernel<<<(E + 255) / 256, 256, 0, stream>>>(ei, deg, E);
  scan_deg_kernel <<<1, 1024, 0, stream>>>(deg, rowptr, cursor, N);
  fill_csr_kernel <<<(E + 255) / 256, 256, 0, stream>>>(ei, cursor, csr, E);

  const int nblocks = (N + ROWS - 1) / ROWS;
  sage_fused_kernel<<<nblocks, 256, 0, stream>>>(x, Wl, bl, Wr, deg, rowptr, csr, out, N);
}